// LBFGSUpdate_52544629899920
// MI455X (gfx1250) — compile-verified
//
#include <hip/hip_runtime.h>
#include <math.h>

#define LH 16  // L-BFGS history length

typedef __attribute__((ext_vector_type(2))) float v2f;
typedef __attribute__((ext_vector_type(8))) float v8f;

// -------- wave(32) + block reduction helper ----------------------------------
__device__ __forceinline__ float wave_red(float v) {
    #pragma unroll
    for (int off = 16; off > 0; off >>= 1) v += __shfl_down(v, off);
    return v;
}

// =============================================================================
// Kernel 1: fused streaming pass over N.
//   accT[h] = sum_n g[h,n]*x[n]          (top, before roll)
//   accB[h] = sum_n y[h,n]*x[n]          (bot, before gamma scale)
//   accC    = sum_n y[idx,n]*g[idx,n]    (curvature)
//   accG    = sum_n g[idx,n]^2
// Writes 34 partials per block; deterministic (no float atomics).
// =============================================================================
__global__ void lbfgs_reduce(const float* __restrict__ x,
                             const float* __restrict__ y,
                             const float* __restrict__ g,
                             const int*   __restrict__ index_start,
                             float* __restrict__ partials,
                             int N) {
    float accT[LH], accB[LH];
    #pragma unroll
    for (int h = 0; h < LH; ++h) { accT[h] = 0.f; accB[h] = 0.f; }
    float accC = 0.f, accG = 0.f;

    const int is  = index_start[0];
    const int idx = ((is - 1) % LH + LH) % LH;

    const long long tid    = (long long)blockIdx.x * blockDim.x + threadIdx.x;
    const long long stride = (long long)gridDim.x * blockDim.x;

    for (long long base = tid * 4; base + 3 < N; base += stride * 4) {
        const float4 xv = *(const float4*)(x + base);
        #pragma unroll
        for (int h = 0; h < LH; ++h) {
            const float4 gv = *(const float4*)(g + (long long)h * N + base);
            const float4 yv = *(const float4*)(y + (long long)h * N + base);
            accT[h] += gv.x*xv.x + gv.y*xv.y + gv.z*xv.z + gv.w*xv.w;
            accB[h] += yv.x*xv.x + yv.y*xv.y + yv.z*xv.z + yv.w*xv.w;
            if (h == idx) {
                accC += yv.x*gv.x + yv.y*gv.y + yv.z*gv.z + yv.w*gv.w;
                accG += gv.x*gv.x + gv.y*gv.y + gv.z*gv.z + gv.w*gv.w;
            }
        }
    }
    // scalar tail (N % 4) handled by one thread
    if (tid == 0) {
        for (int n = N & ~3; n < N; ++n) {
            const float xv = x[n];
            #pragma unroll
            for (int h = 0; h < LH; ++h) {
                const float gv = g[(long long)h * N + n];
                const float yv = y[(long long)h * N + n];
                accT[h] += gv * xv;
                accB[h] += yv * xv;
                if (h == idx) { accC += yv * gv; accG += gv * gv; }
            }
        }
    }

    // block reduce 34 values
    __shared__ float lred[8][34];
    const int lane = threadIdx.x & 31;
    const int wv   = threadIdx.x >> 5;
    float vals[34];
    #pragma unroll
    for (int i = 0; i < LH; ++i) { vals[i] = accT[i]; vals[LH + i] = accB[i]; }
    vals[32] = accC; vals[33] = accG;
    #pragma unroll
    for (int i = 0; i < 34; ++i) {
        const float r = wave_red(vals[i]);
        if (lane == 0) lred[wv][i] = r;
    }
    __syncthreads();
    if (threadIdx.x < 34) {
        float s = 0.f;
        const int nw = blockDim.x >> 5;
        for (int w = 0; w < nw; ++w) s += lred[w][threadIdx.x];
        partials[(long long)blockIdx.x * 34 + threadIdx.x] = s;
    }
}

// =============================================================================
// Kernel 2: single block. Reduce partials; gamma; J^2 via V_WMMA_F32_16X16X4_F32
// (4 issues, K=16); Cholesky; build 32x32 low_tri; two triangular solves with
// the circular roll/unroll; emit gamma + 32 coefficients to res[].
//   res[0] = gamma, res[1..16] = coefA (for G rows), res[17..32] = coefB (for Y)
// =============================================================================
__global__ void lbfgs_solve(const float* __restrict__ partials, int nb,
                            const float* __restrict__ Lmat,
                            const float* __restrict__ diagv,
                            const float* __restrict__ Sin,
                            const int*   __restrict__ index_start,
                            float* __restrict__ res) {
    __shared__ float lred[8][34];
    __shared__ float tot[34];
    __shared__ float Ls[LH * LH];
    __shared__ float dsh[LH];
    __shared__ float sq[LH];
    __shared__ float Jsq[LH * LH];
    __shared__ float T[32 * 32];
    __shared__ float dv[32];
    __shared__ float gam_sh;

    // ---- stage A: reduce per-block partials (deterministic order) ----
    float vals[34];
    #pragma unroll
    for (int i = 0; i < 34; ++i) vals[i] = 0.f;
    for (int b = threadIdx.x; b < nb; b += blockDim.x) {
        #pragma unroll
        for (int i = 0; i < 34; ++i) vals[i] += partials[(long long)b * 34 + i];
    }
    const int lane = threadIdx.x & 31;
    const int wv   = threadIdx.x >> 5;
    #pragma unroll
    for (int i = 0; i < 34; ++i) {
        const float r = wave_red(vals[i]);
        if (lane == 0) lred[wv][i] = r;
    }
    __syncthreads();
    if (threadIdx.x < 34) {
        float s = 0.f;
        const int nw = blockDim.x >> 5;
        for (int w = 0; w < nw; ++w) s += lred[w][threadIdx.x];
        tot[threadIdx.x] = s;
    }
    // ---- stage B: scalars + small matrices into LDS ----
    if (threadIdx.x < LH) {
        float dd = diagv[threadIdx.x];
        dd = (dd == 0.f) ? 1.f : dd;
        dsh[threadIdx.x] = dd;
        sq[threadIdx.x]  = sqrtf(dd);
    }
    for (int i = threadIdx.x; i < LH * LH; i += blockDim.x) Ls[i] = Lmat[i];
    __syncthreads();
    if (threadIdx.x == 0) {
        const float curv = tot[32], gg = tot[33];
        gam_sh = (curv > 0.f) ? (gg / curv) : 1.f;
    }
    __syncthreads();
    const float gamma = gam_sh;

    // ---- stage C: M = (L/d) @ L^T via WMMA f32 16x16x4, K=16 in 4 chunks ----
    // A layout (ISA 7.12.2): M = lane&15; K = kc + vgpr + 2*(lane>=16)
    // B layout (mirrored) : N = lane&15; K = kc + vgpr + 2*(lane>=16); B[k][n]=L[n][k]
    // Executed uniformly by every wave (EXEC must be all ones for WMMA).
    const int mA = lane & 15;
    const int hi = (lane >> 4) & 1;
    v8f c = {0.f, 0.f, 0.f, 0.f, 0.f, 0.f, 0.f, 0.f};
    #pragma unroll
    for (int kc = 0; kc < LH; kc += 4) {
        const int k0 = kc + 2 * hi;
        v2f a, b;
        a.x = Ls[mA * LH + k0]     / dsh[k0];
        a.y = Ls[mA * LH + k0 + 1] / dsh[k0 + 1];
        b.x = Ls[mA * LH + k0];       // B[k][n] = L[n][k], n == lane&15 == mA
        b.y = Ls[mA * LH + k0 + 1];
        c = __builtin_amdgcn_wmma_f32_16x16x4_f32(false, a, false, b,
                                                  (short)0, c, false, false);
    }
    if (threadIdx.x < 32) {   // wave 0 writes (all waves hold identical D)
        #pragma unroll
        for (int r = 0; r < 8; ++r) {
            const int m = r + 8 * hi;
            const int n = lane & 15;
            Jsq[m * LH + n] = gamma * Sin[m * LH + n] + c[r];
        }
    }
    __syncthreads();

    // ---- stage D: Cholesky, build low_tri, rolls + two triangular solves ----
    if (threadIdx.x == 0) {
        // Cholesky in-place (lower) of Jsq
        for (int i = 0; i < LH; ++i) {
            for (int j = 0; j <= i; ++j) {
                float s = Jsq[i * LH + j];
                for (int k = 0; k < j; ++k) s -= Jsq[i * LH + k] * Jsq[j * LH + k];
                Jsq[i * LH + j] = (i == j) ? sqrtf(s) : (s / Jsq[j * LH + j]);
            }
        }
        // low_tri = [[diag(sqrt_d), 0], [-L/sqrt_d, J]]
        for (int i = 0; i < 32 * 32; ++i) T[i] = 0.f;
        for (int i = 0; i < LH; ++i) T[i * 32 + i] = sq[i];
        for (int i = 0; i < LH; ++i)
            for (int j = 0; j < LH; ++j) {
                T[(LH + i) * 32 + j] = -Ls[i * LH + j] / sq[j];
                if (j <= i) T[(LH + i) * 32 + LH + j] = Jsq[i * LH + j];
            }
        // descent with circular roll (active when index_start > H)
        const int is   = index_start[0];
        const int roll = (is > LH) ? (((is % LH) + LH) % LH) : 0;
        for (int i = 0; i < LH; ++i) {
            dv[i]      = tot[(i + roll) % LH];                 // top
            dv[LH + i] = gamma * tot[LH + (i + roll) % LH];    // bot
        }
        // forward solve (lower)
        for (int i = 0; i < 32; ++i) {
            float s = dv[i];
            for (int j = 0; j < i; ++j) s -= T[i * 32 + j] * dv[j];
            dv[i] = s / T[i * 32 + i];
        }
        // backward solve: U = T^T with first H rows negated
        for (int i = 31; i >= 0; --i) {
            float s = dv[i];
            for (int j = i + 1; j < 32; ++j) {
                float u = T[j * 32 + i];
                if (i < LH) u = -u;
                s -= u * dv[j];
            }
            float uii = T[i * 32 + i];
            if (i < LH) uii = -uii;
            dv[i] = s / uii;
        }
        // un-roll and emit coefficients
        res[0] = gamma;
        for (int i = 0; i < LH; ++i) {
            const int src = ((i - roll) % LH + LH) % LH;
            res[1 + i]  = dv[src];                 // multiplies g rows
            res[17 + i] = gamma * dv[LH + src];    // multiplies y rows
        }
    }
}

// =============================================================================
// Kernel 3: out[n] = gamma*x[n] - sum_h coefA[h]*g[h,n] - sum_h coefB[h]*y[h,n]
// =============================================================================
__global__ void lbfgs_combine(const float* __restrict__ x,
                              const float* __restrict__ y,
                              const float* __restrict__ g,
                              const float* __restrict__ res,
                              float* __restrict__ out, int N) {
    __shared__ float cA[LH], cB[LH], gsh;
    if (threadIdx.x == 0) gsh = res[0];
    if (threadIdx.x < LH) {
        cA[threadIdx.x] = res[1 + threadIdx.x];
        cB[threadIdx.x] = res[17 + threadIdx.x];
    }
    __syncthreads();
    const float gamma = gsh;

    const long long tid    = (long long)blockIdx.x * blockDim.x + threadIdx.x;
    const long long stride = (long long)gridDim.x * blockDim.x;
    for (long long base = tid * 4; base + 3 < N; base += stride * 4) {
        const float4 xv = *(const float4*)(x + base);
        float4 acc;
        acc.x = gamma * xv.x; acc.y = gamma * xv.y;
        acc.z = gamma * xv.z; acc.w = gamma * xv.w;
        #pragma unroll
        for (int h = 0; h < LH; ++h) {
            const float4 gv = *(const float4*)(g + (long long)h * N + base);
            const float4 yv = *(const float4*)(y + (long long)h * N + base);
            const float a = cA[h], b = cB[h];
            acc.x -= a * gv.x + b * yv.x;
            acc.y -= a * gv.y + b * yv.y;
            acc.z -= a * gv.z + b * yv.z;
            acc.w -= a * gv.w + b * yv.w;
        }
        *(float4*)(out + base) = acc;
    }
    if (tid == 0) {
        for (int n = N & ~3; n < N; ++n) {
            float acc = gamma * x[n];
            #pragma unroll
            for (int h = 0; h < LH; ++h)
                acc -= cA[h] * g[(long long)h * N + n] + cB[h] * y[(long long)h * N + n];
            out[n] = acc;
        }
    }
}

// =============================================================================
extern "C" void kernel_launch(void* const* d_in, const int* in_sizes, int n_in,
                              void* d_out, int out_size, void* d_ws, size_t ws_size,
                              hipStream_t stream) {
    const float* x    = (const float*)d_in[0];   // pytree            [N]
    const float* y    = (const float*)d_in[1];   // y_diff_history    [H,N]
    const float* g    = (const float*)d_in[2];   // grad_diff_history [H,N]
    const float* Lm   = (const float*)d_in[3];   // cross (L)         [H,H]
    const float* dgv  = (const float*)d_in[4];   // diag              [H]
    const float* Sin  = (const float*)d_in[5];   // S^T S             [H,H]
    const int*   istt = (const int*)d_in[6];     // index_start       [1]
    float*       out  = (float*)d_out;
    const int    N    = in_sizes[0];

    float* wsf      = (float*)d_ws;
    float* res      = wsf;         // 33 floats
    float* partials = wsf + 64;    // nb * 34 floats

    // clamp block count to workspace capacity (deterministic: ws_size is fixed)
    long long maxnb = ((long long)(ws_size / sizeof(float)) - 64) / 34;
    int nb = 2048;
    if (maxnb < nb) nb = (int)maxnb;
    if (nb < 1) nb = 1;

    lbfgs_reduce <<<nb,   256, 0, stream>>>(x, y, g, istt, partials, N);
    lbfgs_solve  <<<1,    256, 0, stream>>>(partials, nb, Lm, dgv, Sin, istt, res);
    lbfgs_combine<<<2048, 256, 0, stream>>>(x, y, g, res, out, N);
}